// PatchMasker_15556371546170
// MI455X (gfx1250) — compile-verified
//
#include <hip/hip_runtime.h>

// PatchMasker: out_k = mask[t] ? 0.0f : in_k  for three (B,T,F) f32 tensors.
// HBM-bound streaming op (0 FLOPs/byte) -> optimize the memory path:
//   - one wave32 == one (b,t) row (1 KiB/tensor): mask is wave-uniform by
//     construction, so readfirstlane -> scalar mask load + scalar branch
//   - 128-bit NT loads/stores (read-once/write-once, working set >> 192MB L2)
//   - masked rows skip all loads (saves 40% of read traffic)

typedef float v4f __attribute__((ext_vector_type(4)));

static constexpr int       kB    = 256;
static constexpr int       kT    = 512;
static constexpr int       kF    = 256;
static constexpr long long kN    = (long long)kB * kT * kF;  // elems per tensor
static constexpr long long kNV   = kN / 4;                   // vec4 per tensor
static constexpr long long kRows = (long long)kB * kT;       // 131072 rows
// 64 vec4 per row; 32 lanes/wave -> 2 vec4 per lane, wave covers one row.

__global__ __launch_bounds__(256)
void patch_mask_kernel(const v4f* __restrict__ xd,
                       const v4f* __restrict__ xt,
                       const v4f* __restrict__ xs,
                       const int* __restrict__ mask,   // (T,) 0/1
                       v4f* __restrict__ zd,
                       v4f* __restrict__ zt,
                       v4f* __restrict__ zs)
{
    const long long tid  = (long long)blockIdx.x * blockDim.x + threadIdx.x;
    const long long row  = tid >> 5;              // one wave32 per row
    const int       lane = threadIdx.x & 31;
    if (row >= kRows) return;

    // t is identical across the wave (row = tid>>5); pin it to an SGPR so the
    // mask read becomes s_load_b32 and the branch is a scalar s_cbranch.
    const int t = __builtin_amdgcn_readfirstlane((int)(row & (kT - 1)));

    const long long v = row * 64 + (long long)lane * 2;  // first vec4 index

    if (mask[t] != 0) {
        // Masked timestep: no loads, stream zeros (write-once -> NT).
        const v4f z = (v4f)0.0f;
        __builtin_nontemporal_store(z, &zd[v]);
        __builtin_nontemporal_store(z, &zd[v + 1]);
        __builtin_nontemporal_store(z, &zt[v]);
        __builtin_nontemporal_store(z, &zt[v + 1]);
        __builtin_nontemporal_store(z, &zs[v]);
        __builtin_nontemporal_store(z, &zs[v + 1]);
    } else {
        const v4f a0 = __builtin_nontemporal_load(&xd[v]);
        const v4f a1 = __builtin_nontemporal_load(&xd[v + 1]);
        const v4f b0 = __builtin_nontemporal_load(&xt[v]);
        const v4f b1 = __builtin_nontemporal_load(&xt[v + 1]);
        const v4f c0 = __builtin_nontemporal_load(&xs[v]);
        const v4f c1 = __builtin_nontemporal_load(&xs[v + 1]);
        __builtin_nontemporal_store(a0, &zd[v]);
        __builtin_nontemporal_store(a1, &zd[v + 1]);
        __builtin_nontemporal_store(b0, &zt[v]);
        __builtin_nontemporal_store(b1, &zt[v + 1]);
        __builtin_nontemporal_store(c0, &zs[v]);
        __builtin_nontemporal_store(c1, &zs[v + 1]);
    }
}

extern "C" void kernel_launch(void* const* d_in, const int* in_sizes, int n_in,
                              void* d_out, int out_size, void* d_ws, size_t ws_size,
                              hipStream_t stream)
{
    (void)in_sizes; (void)n_in; (void)out_size; (void)d_ws; (void)ws_size;

    const v4f* xd   = (const v4f*)d_in[0];
    const v4f* xt   = (const v4f*)d_in[1];
    const v4f* xs   = (const v4f*)d_in[2];
    const int* mask = (const int*)d_in[3];

    v4f* zd = (v4f*)d_out;
    v4f* zt = zd + kNV;
    v4f* zs = zd + 2 * kNV;

    const int threads = 256;                                   // 8 wave32/block
    const long long totalThreads = kRows * 32;                 // 4,194,304
    const long long blocks = (totalThreads + threads - 1) / threads; // 16384

    patch_mask_kernel<<<dim3((unsigned)blocks), dim3(threads), 0, stream>>>(
        xd, xt, xs, mask, zd, zt, zs);
}